// ShiftLayer_66932770341347
// MI455X (gfx1250) — compile-verified
//
#include <hip/hip_runtime.h>

// Shift layer: out[b,t,c] = in[b, t + off[c%3], c], off = {0,-1,+1}, zero-padded.
// [B,L,C] = [8,8192,384] f32. Pure streaming: ~201 MB total -> ~8.6us floor @ 23.3 TB/s.
//
// Strategy (MI455X): stage TT+2 full rows into LDS via GLOBAL_LOAD_ASYNC_TO_LDS_B128
// (128b coalesced, ASYNCcnt-tracked, no VGPR round trip), s_wait_asynccnt + barrier,
// resolve the per-channel shift as a conflict-free LDS gather (row pitch 384 words
// = 0 mod 64 banks -> bank depends only on channel), then stream coalesced 128b
// nontemporal stores. All div/mod hoisted out of the hot loops.

#define BB 8
#define LL 8192
#define CC 384
#define TT 32                  // time steps per workgroup tile
#define ROWS (TT + 2)          // staged rows incl. halo
#define ROW_F4 (CC / 4)        // 96 float4 per row
#define NTHREADS 384           // 4 row-groups x 96 columns, 12 wave32s

typedef __attribute__((ext_vector_type(4))) float v4f;
typedef __attribute__((ext_vector_type(4))) int   v4i;

#if defined(__gfx1250__) && __has_builtin(__builtin_amdgcn_global_load_async_to_lds_b128)
#define HAVE_ASYNC_LDS 1
typedef __attribute__((address_space(1))) v4i g_v4i;   // global (AS1) int4
typedef __attribute__((address_space(3))) v4i l_v4i;   // LDS (AS3) int4
#else
#define HAVE_ASYNC_LDS 0
#endif

__device__ __forceinline__ int shift_words(int ch) {
    int m = ch % 3;                    // 0 -> same row, 1 -> t-1, 2 -> t+1
    return (m == 0) ? 0 : ((m == 1) ? -CC : CC);
}

__global__ __launch_bounds__(NTHREADS)
void shift_layer_kernel(const float* __restrict__ in, float* __restrict__ out) {
    __shared__ float smem[ROWS * CC];

    const int tilesPerB = LL / TT;
    const int b  = blockIdx.x / tilesPerB;
    const int t0 = (blockIdx.x % tilesPerB) * TT;

    // One-time per-thread decomposition: fixed column, strided rows.
    const int c4 = threadIdx.x % ROW_F4;   // float4 column 0..95
    const int rg = threadIdx.x / ROW_F4;   // row group 0..3
    const int c  = c4 * 4;                 // base channel of this thread's float4

    const float* __restrict__ inb = in + ((size_t)b * LL) * CC;

    // ---- Stage 1: async b128 global -> LDS for rows [t0-1, t0+TT] ----
    for (int r = rg; r < ROWS; r += 4) {
        const int g = t0 - 1 + r;                  // global row (halo may be OOB)
        float* lp = &smem[r * CC + c];
        if (g >= 0 && g < LL) {
            const float* gp = inb + (size_t)g * CC + c;
#if HAVE_ASYNC_LDS
            __builtin_amdgcn_global_load_async_to_lds_b128(
                (g_v4i*)gp, (l_v4i*)lp, /*offset=*/0, /*cpol=*/0);
#else
            *(v4f*)lp = *(const v4f*)gp;
#endif
        } else {
            v4f z = {0.f, 0.f, 0.f, 0.f};          // zero padding rows
            *(v4f*)lp = z;
        }
    }

#if HAVE_ASYNC_LDS
#if __has_builtin(__builtin_amdgcn_s_wait_asynccnt)
    __builtin_amdgcn_s_wait_asynccnt(0);
#else
    asm volatile("s_wait_asynccnt 0" ::: "memory");
#endif
#endif
    __syncthreads();

    // ---- Stage 2: gather shifted channels from LDS, stream b128 stores ----
    // Shift offsets are invariant along t: compute once per thread.
    const int o0 = c + 0 + shift_words(c + 0);
    const int o1 = c + 1 + shift_words(c + 1);
    const int o2 = c + 2 + shift_words(c + 2);
    const int o3 = c + 3 + shift_words(c + 3);

    float* __restrict__ outb = out + ((size_t)b * LL + t0) * CC + c;

    for (int tl = rg; tl < TT; tl += 4) {
        const float* base = &smem[(tl + 1) * CC];
        v4f v;
        v.x = base[o0];
        v.y = base[o1];
        v.z = base[o2];
        v.w = base[o3];
        __builtin_nontemporal_store(v, (v4f*)(outb + (size_t)tl * CC));
    }
}

extern "C" void kernel_launch(void* const* d_in, const int* in_sizes, int n_in,
                              void* d_out, int out_size, void* d_ws, size_t ws_size,
                              hipStream_t stream) {
    const float* in = (const float*)d_in[0];
    float* out = (float*)d_out;
    const int nblocks = BB * (LL / TT);   // 2048 workgroups, one (b, t-tile) each
    shift_layer_kernel<<<nblocks, NTHREADS, 0, stream>>>(in, out);
}